// SPLoss_1958505087536
// MI455X (gfx1250) — compile-verified
//
#include <hip/hip_runtime.h>
#include <hip/hip_bf16.h>
#include <float.h>

// ---------------- configuration ----------------
#define CDIM            1000           // class count (fixed by reference)
#define ROWS_PER_TILE   8              // one row per wave, 8 waves / block
#define TILE_BYTES      (ROWS_PER_TILE * CDIM * 4)
#define BLK1            256            // 8 wave32 waves
#define BLK2            1024

#if defined(__HIP_DEVICE_COMPILE__) && defined(__gfx1250__) && \
    __has_builtin(__builtin_amdgcn_tensor_load_to_lds) && \
    __has_builtin(__builtin_amdgcn_s_wait_tensorcnt)
#define USE_TDM 1
#else
#define USE_TDM 0
#endif

#if USE_TDM
typedef unsigned int u32x4 __attribute__((ext_vector_type(4)));
typedef int          i32x8 __attribute__((ext_vector_type(8)));
typedef int          i32x4 __attribute__((ext_vector_type(4)));

// Issue one TDM load: tile of ROWS_PER_TILE rows x CDIM fp32 into LDS at the byte address in lds_off.
// Descriptor per CDNA5 ISA section 8 (D# groups 0..3).
// clang-23 toolchain builtin arity: (v4u g0, v8i g1, v4i g2, v4i g3, v8i g4, i32 cpol)
__device__ __forceinline__ void tdm_issue_tile(const float* src, int tile,
                                               unsigned lds_off, int B) {
  unsigned long long ga = (unsigned long long)(uintptr_t)src +
                          (unsigned long long)tile * TILE_BYTES;
  u32x4 g0;
  g0[0] = 1u;                                   // count=1 (valid), user mode
  g0[1] = lds_off;                              // lds_addr (bytes)
  g0[2] = (unsigned)ga;                         // global_addr[31:0]
  g0[3] = ((unsigned)(ga >> 32) & 0x01FFFFFFu)  // global_addr[56:32]
          | 0x80000000u;                        // type=2 ("image") @ bits127:126
  i32x8 g1;
  g1[0] = 0x00020000;                           // data_size=2 (4 bytes), no flags
  g1[1] = (int)(((unsigned)CDIM & 0xFFFFu) << 16);                 // tensor_dim0 lo16
  g1[2] = (int)((((unsigned)CDIM >> 16) & 0xFFFFu) |
                (((unsigned)B & 0xFFFFu) << 16));                  // dim0 hi16 | dim1 lo16
  g1[3] = (int)((((unsigned)B >> 16) & 0xFFFFu) |
                (((unsigned)CDIM & 0xFFFFu) << 16));               // dim1 hi16 | tile_dim0
  g1[4] = ROWS_PER_TILE;                        // tile_dim1 (tile_dim2=0)
  g1[5] = CDIM;                                 // tensor_dim0_stride lo32
  g1[6] = 0;                                    // stride hi | dim1_stride lo
  g1[7] = 0;
  i32x4 g2; g2[0] = 1; g2[1] = 1; g2[2] = 0; g2[3] = 0;  // benign higher dims
  i32x4 g3; g3[0] = 0; g3[1] = 0; g3[2] = 0; g3[3] = 0;
  i32x8 g4 = {0, 0, 0, 0, 0, 0, 0, 0};                   // zero-filled (probe form)
  __builtin_amdgcn_tensor_load_to_lds(g0, g1, g2, g3, g4, 0);
}
#endif

// One wave computes CE for one row (CDIM=1000 floats at `row`).
__device__ __forceinline__ void process_row(const float* __restrict__ row, int r,
                                            const int* __restrict__ tgt,
                                            float* __restrict__ ce,
                                            int lane, int B) {
  if (r >= B) return;                            // wave-uniform
  const float4* rp = (const float4*)row;
  float4 v[8];
#pragma unroll
  for (int c = 0; c < 7; ++c) v[c] = rp[c * 32 + lane];   // 896 elems
  // tail: elements 896..999 -> lanes 0..25 (26 float4s), pad rest
  if (896 + lane * 4 < CDIM) v[7] = rp[224 + lane];
  else v[7] = make_float4(-FLT_MAX, -FLT_MAX, -FLT_MAX, -FLT_MAX);

  float m = -FLT_MAX;
#pragma unroll
  for (int c = 0; c < 8; ++c)
    m = fmaxf(m, fmaxf(fmaxf(v[c].x, v[c].y), fmaxf(v[c].z, v[c].w)));
#pragma unroll
  for (int o = 16; o; o >>= 1) m = fmaxf(m, __shfl_xor(m, o, 32));

  const float L2E = 1.4426950408889634f;
  float s = 0.f;
#pragma unroll
  for (int c = 0; c < 8; ++c) {
    s += exp2f((v[c].x - m) * L2E);  // pad lanes: (-FLT_MAX-m)*L2E -> -inf -> 0
    s += exp2f((v[c].y - m) * L2E);
    s += exp2f((v[c].z - m) * L2E);
    s += exp2f((v[c].w - m) * L2E);
  }
#pragma unroll
  for (int o = 16; o; o >>= 1) s += __shfl_xor(s, o, 32);

  if (lane == 0) {
    int ti = tgt[r];
    float xt = row[ti];
    ce[r] = m + logf(s) - xt;        // -(x_t - max - log(sum))
  }
}

// Phase 1: per-row cross entropy. TDM double-buffered global->LDS streaming.
__global__ void __launch_bounds__(BLK1)
ce_kernel(const float* __restrict__ logits, const int* __restrict__ tgt,
          float* __restrict__ ce, int B, int tiles) {
  const int lane = threadIdx.x & 31;
  const int wave = threadIdx.x >> 5;
#if USE_TDM
  __shared__ __align__(16) float buf[2][ROWS_PER_TILE][CDIM];   // 64 000 B
  const int stride = gridDim.x;
  // prologue: stage first tile into buffer 0
  if (threadIdx.x < 32) {
    int t0 = blockIdx.x;
    if (t0 < tiles)
      tdm_issue_tile(logits, t0, (unsigned)(uintptr_t)&buf[0][0][0], B);
  }
  int iter = 0;
  for (int t = blockIdx.x; t < tiles; t += stride, ++iter) {
    if (threadIdx.x < 32) {
      int nt = t + stride;
      if (nt < tiles) {
        tdm_issue_tile(logits, nt,
                       (unsigned)(uintptr_t)&buf[(iter + 1) & 1][0][0], B);
        __builtin_amdgcn_s_wait_tensorcnt(1);    // current tile resident
      } else {
        __builtin_amdgcn_s_wait_tensorcnt(0);
      }
    }
    __syncthreads();                             // publish LDS tile to all waves
    process_row(&buf[iter & 1][wave][0], t * ROWS_PER_TILE + wave,
                tgt, ce, lane, B);
    __syncthreads();                             // buffer reuse fence
  }
#else
  for (int r = blockIdx.x * ROWS_PER_TILE + wave; r < B;
       r += gridDim.x * ROWS_PER_TILE)
    process_row(logits + (size_t)r * CDIM, r, tgt, ce, lane, B);
#endif
}

// order-preserving uint key for floats (handles any sign robustly)
__device__ __forceinline__ unsigned fkey(float x) {
  unsigned u = __float_as_uint(x);
  return (u & 0x80000000u) ? ~u : (u | 0x80000000u);
}

// Phase 2+3: exact k-th order statistic via 4x8-bit radix select, then
// deterministic tree-reduced weighted mean. Single workgroup (B = 32K is tiny).
__global__ void __launch_bounds__(BLK2)
select_reduce(const float* __restrict__ ce, const int* __restrict__ epoch_p,
              float* __restrict__ out, int B) {
  __shared__ unsigned hist[256];
  __shared__ unsigned sh_prefix;
  __shared__ int sh_k;
  __shared__ float red[BLK2];
  const int tid = threadIdx.x;

  const int epoch = epoch_p[0];
  float ratio; int q;
  if (epoch <= 25)      { ratio = 0.7f; q = 2; }
  else if (epoch <= 75) { ratio = 0.8f; q = 0; }
  else if (epoch <= 95) { ratio = 0.9f; q = 0; }
  else                  { ratio = 1.0f; q = 0; }
  int k = (int)((float)B * ratio);
  if (k >= B) k = B - 1;                 // JAX clamps OOB index

  if (tid == 0) { sh_prefix = 0u; sh_k = k; }
  __syncthreads();

#pragma unroll
  for (int pass = 0; pass < 4; ++pass) {
    const int shift = 24 - pass * 8;
    if (tid < 256) hist[tid] = 0u;
    __syncthreads();
    const unsigned pref = sh_prefix;
    for (int i = tid; i < B; i += BLK2) {
      unsigned key = fkey(ce[i]);
      bool match = (pass == 0) || ((key >> (shift + 8)) == (pref >> (shift + 8)));
      if (match) atomicAdd(&hist[(key >> shift) & 0xFFu], 1u);
    }
    __syncthreads();
    if (tid == 0) {
      int kk = sh_k; unsigned acc = 0; int d = 0;
      for (; d < 256; ++d) { unsigned c = hist[d]; if (acc + c > (unsigned)kk) break; acc += c; }
      sh_k = kk - (int)acc;
      sh_prefix = pref | ((unsigned)d << shift);
    }
    __syncthreads();
  }

  const unsigned key = sh_prefix;
  const unsigned u = (key & 0x80000000u) ? (key ^ 0x80000000u) : ~key;
  const float lam = __uint_as_float(u);

  float acc = 0.f;
  for (int i = tid; i < B; i += BLK2) {
    float c = ce[i];
    float v;
    if (c >= lam) v = 0.f;
    else {
      float base = 1.f - c / lam;          // > 0 here
      v = (q == 2) ? base : 1.f / base;    // exponent = 1/(q-1) in {1,-1}
      if (v >= 10.f) v = 10.f;
    }
    acc += c * v;
  }
  red[tid] = acc;
  __syncthreads();
  for (int s = BLK2 / 2; s > 0; s >>= 1) {
    if (tid < s) red[tid] += red[tid + s];
    __syncthreads();
  }
  if (tid == 0) out[0] = red[0] / (float)B;
}

extern "C" void kernel_launch(void* const* d_in, const int* in_sizes, int n_in,
                              void* d_out, int out_size, void* d_ws, size_t ws_size,
                              hipStream_t stream) {
  const float* logits = (const float*)d_in[0];
  const int*   tgt    = (const int*)d_in[1];
  const int*   epoch  = (const int*)d_in[2];
  const int B = in_sizes[1];                          // 32768
  float* ce = (float*)d_ws;                           // B floats scratch
  float* out = (float*)d_out;

  const int tiles = (B + ROWS_PER_TILE - 1) / ROWS_PER_TILE;   // 4096
  int blocks = tiles < 1024 ? tiles : 1024;
  ce_kernel<<<blocks, BLK1, 0, stream>>>(logits, tgt, ce, B, tiles);
  select_reduce<<<1, BLK2, 0, stream>>>(ce, epoch, out, B);
}